// DifferentiableFBP_77824807404260
// MI455X (gfx1250) — compile-verified
//
#include <hip/hip_runtime.h>
#include <math.h>

typedef float v2f __attribute__((ext_vector_type(2)));
typedef float v8f __attribute__((ext_vector_type(8)));

#define B_   2
#define A_   1152
#define DET_ 736
#define N_   512
#define PI_  3.14159265358979323846f
#define CHUNKA 8

// Workspace layout (float offsets)
#define OFF_SCAL 0                          // 16 per batch: [dso_s, du_v, dbeta, beta0, hu0]
#define OFF_TRIG 32                         // per batch 2304: sin[1152] then cos[1152]
#define OFF_COSW (32 + B_*2304)             // per batch 736
#define OFF_T    (OFF_COSW + B_*DET_)       // 736x736 symmetric Toeplitz ramp matrix
#define OFF_WSINO (OFF_T + DET_*DET_)       // B*1152*736 weighted sinogram
#define OFF_FILT  (OFF_WSINO + B_*A_*DET_)  // B*1152*736 filtered sinogram

// ---------------------------------------------------------------------------
// Kernel 1: per-batch scalars, sin/cos table, cos-weight table
// ---------------------------------------------------------------------------
__global__ void fbp_prep(const float* __restrict__ angles, const float* __restrict__ dso,
                         const float* __restrict__ ddo, const float* __restrict__ du,
                         const float* __restrict__ hu, float* __restrict__ ws) {
    const int b = blockIdx.x;
    const int tid = threadIdx.x;
    const float vox  = 1.0f / 0.7f;
    const float dso_s = vox * dso[b];
    const float sd_s  = vox * (dso[b] + ddo[b]);
    const float du_s  = vox * du[b];
    const float du_v  = du_s * dso_s / sd_s;
    const float beta0 = angles[b * A_ + 0];
    const float inc   = angles[b * A_ + 1] - beta0;
    const float dbeta = ((float)A_ * inc) / (float)A_;

    if (tid == 0) {
        float* s = ws + OFF_SCAL + b * 16;
        s[0] = dso_s; s[1] = du_v; s[2] = dbeta; s[3] = beta0;
        s[4] = fmaxf(fabsf(hu[b]), 1e-6f);
    }
    float* tr = ws + OFF_TRIG + b * 2304;
    for (int a = tid; a < A_; a += blockDim.x) {
        float beta = beta0 + dbeta * (float)a;
        tr[a]      = sinf(beta);
        tr[A_ + a] = cosf(beta);
    }
    float* cw = ws + OFF_COSW + b * DET_;
    const float center = (float)(DET_ - 1) * 0.5f;
    for (int j = tid; j < DET_; j += blockDim.x) {
        float uk = ((float)j - center) * du_v;
        cw[j] = dso_s / sqrtf(dso_s * dso_s + uk * uk);
    }
}

// ---------------------------------------------------------------------------
// Kernel 1b: materialize the symmetric Toeplitz ramp matrix T[i,j] = h(|i-j|)
// h(0)=0.5, h(odd m)=-2/(pi m)^2, h(even m!=0)=0.  2.2 MB -> lives in L2.
// ---------------------------------------------------------------------------
__global__ void fbp_toeplitz(float* __restrict__ ws) {
    int idx = blockIdx.x * blockDim.x + threadIdx.x;
    if (idx >= DET_ * DET_) return;
    int i = idx / DET_, j = idx % DET_;
    int m = i - j; if (m < 0) m = -m;
    float v = 0.0f;
    if (m == 0)     v = 0.5f;
    else if (m & 1) { float fm = (float)m; v = -2.0f / (PI_ * PI_ * fm * fm); }
    ws[OFF_T + idx] = v;
}

// ---------------------------------------------------------------------------
// Kernel 2: cosine-weight the sinogram (sino * vox * cosw[j])
// ---------------------------------------------------------------------------
__global__ void fbp_weight(const float* __restrict__ sino, float* __restrict__ ws) {
    int idx = blockIdx.x * blockDim.x + threadIdx.x;
    const int total = B_ * A_ * DET_;
    if (idx >= total) return;
    int b = idx / (A_ * DET_);
    int j = idx % DET_;
    const float vox = 1.0f / 0.7f;
    ws[OFF_WSINO + idx] = sino[idx] * vox * ws[OFF_COSW + b * DET_ + j];
}

// ---------------------------------------------------------------------------
// Kernel 3: ramp filter as GEMM vs. precomputed Toeplitz T via
// V_WMMA_F32_16X16X4_F32.  Each wave computes TWO 16(angle)x16(det) output
// tiles, reusing the A registers for both; B-tiles are contiguous b64 loads
// from T (T symmetric: B[k,n] = T[col*736 + k]).  No LDS, no gather math.
// A layout (16x4 f32): lanes 0-15 hold K=k0+0,k0+1 ; lanes 16-31 K=k0+2,k0+3.
// ---------------------------------------------------------------------------
__global__ void __launch_bounds__(256) fbp_filter_wmma(float* __restrict__ ws) {
    const int wave = threadIdx.x >> 5;
    const int lane = threadIdx.x & 31;
    const int slot = blockIdx.x * 8 + wave;      // 24 slots cover 23 pairs of col tiles
    if (slot >= (DET_ / 16 + 1) / 2) return;     // 46 col tiles -> 23 pairs
    const int rowTile = blockIdx.y;
    const int b = blockIdx.z;

    const float inv_du_v = 1.0f / ws[OFF_SCAL + b * 16 + 1];
    const float* Abase = ws + OFF_WSINO + (size_t)b * A_ * DET_
                            + (size_t)(rowTile * 16) * DET_;
    const float* Tm = ws + OFF_T;

    const int n    = lane & 15;
    const int koff = (lane >> 4) * 2;            // 0 for lanes 0-15, 2 for lanes 16-31
    const int col0 = slot * 32 + n;
    const int col1 = col0 + 16;

    const float* ap  = Abase + (size_t)(lane & 15) * DET_ + koff;
    const float* bp0 = Tm + (size_t)col0 * DET_ + koff;
    const float* bp1 = Tm + (size_t)col1 * DET_ + koff;

    v8f c0 = {};
    v8f c1 = {};
    for (int k0 = 0; k0 < DET_; k0 += 4) {
        v2f a  = *(const v2f*)(ap  + k0);
        v2f b0 = *(const v2f*)(bp0 + k0);
        v2f b1 = *(const v2f*)(bp1 + k0);
        c0 = __builtin_amdgcn_wmma_f32_16x16x4_f32(false, a, false, b0,
                                                   (short)0, c0, false, false);
        c1 = __builtin_amdgcn_wmma_f32_16x16x4_f32(false, a, false, b1,
                                                   (short)0, c1, false, false);
    }

    float* Dbase = ws + OFF_FILT + (size_t)b * A_ * DET_;
    const int mbase = (lane < 16) ? 0 : 8;       // D layout: lanes 16-31 hold M=8..15
#pragma unroll
    for (int v = 0; v < 8; ++v) {
        size_t row = (size_t)(rowTile * 16 + mbase + v) * DET_;
        Dbase[row + col0] = c0[v] * inv_du_v;
        Dbase[row + col1] = c1[v] * inv_du_v;
    }
}

// ---------------------------------------------------------------------------
// Kernel 4: fan-beam backprojection + HU normalization.
// 16x16 pixel tile per block; 8 angle rows (23.5 KB) staged in LDS per barrier.
// ---------------------------------------------------------------------------
__global__ void __launch_bounds__(256) fbp_backproject(const float* __restrict__ ws,
                                                       float* __restrict__ out) {
    __shared__ float rows[CHUNKA * DET_];
    __shared__ float s_sin[CHUNKA];
    __shared__ float s_cos[CHUNKA];

    const int b = blockIdx.z;
    const float dso_s = ws[OFF_SCAL + b * 16 + 0];
    const float du_v  = ws[OFF_SCAL + b * 16 + 1];
    const float dbeta = ws[OFF_SCAL + b * 16 + 2];
    const float hu0   = ws[OFF_SCAL + b * 16 + 4];
    const float inv_du_v = 1.0f / du_v;
    const float center = (float)(DET_ - 1) * 0.5f;

    const int px = blockIdx.x * 16 + threadIdx.x;
    const int py = blockIdx.y * 16 + threadIdx.y;
    const float X = (float)px - (float)(N_ - 1) * 0.5f;
    const float Y = (float)py - (float)(N_ - 1) * 0.5f;
    const int tid = threadIdx.y * 16 + threadIdx.x;

    const float* fbase = ws + OFF_FILT + (size_t)b * A_ * DET_;
    const float* trig  = ws + OFF_TRIG + b * 2304;

    float acc = 0.0f;
    for (int a0 = 0; a0 < A_; a0 += CHUNKA) {
        __syncthreads();
        for (int i = tid; i < CHUNKA * DET_; i += 256)
            rows[i] = fbase[(size_t)a0 * DET_ + i];
        if (tid < CHUNKA) {
            s_sin[tid] = trig[a0 + tid];
            s_cos[tid] = trig[A_ + a0 + tid];
        }
        __syncthreads();
        if (a0 + CHUNKA < A_)   // hide L2 latency of the next chunk
            __builtin_prefetch(fbase + (size_t)(a0 + CHUNKA) * DET_ + tid * 8, 0, 1);
#pragma unroll
        for (int aa = 0; aa < CHUNKA; ++aa) {
            float sb = s_sin[aa], cb = s_cos[aa];
            float U  = dso_s + X * sb - Y * cb;
            float rU = 1.0f / U;
            float u  = dso_s * (X * cb + Y * sb) * rU * inv_du_v + center;
            float i0f = floorf(u);
            float w   = u - i0f;
            int i0 = (int)i0f;
            int c0 = min(max(i0, 0), DET_ - 1);
            int c1 = min(max(i0 + 1, 0), DET_ - 1);
            float q0 = rows[aa * DET_ + c0];
            float q1 = rows[aa * DET_ + c1];
            float r  = dso_s * rU;
            float contrib = (q0 + (q1 - q0) * w) * r * r;
            if (u >= 0.0f && u <= (float)(DET_ - 1)) acc += contrib;
        }
    }
    float reco = acc * 0.5f * dbeta;
    out[(size_t)b * N_ * N_ + (size_t)py * N_ + px] =
        1000.0f * (reco - hu0) / (hu0 + 1e-6f);
}

// ---------------------------------------------------------------------------
extern "C" void kernel_launch(void* const* d_in, const int* in_sizes, int n_in,
                              void* d_out, int out_size, void* d_ws, size_t ws_size,
                              hipStream_t stream) {
    (void)in_sizes; (void)n_in; (void)out_size; (void)ws_size;
    const float* sino   = (const float*)d_in[0];
    const float* angles = (const float*)d_in[1];
    const float* dso    = (const float*)d_in[2];
    const float* ddo    = (const float*)d_in[3];
    const float* du     = (const float*)d_in[4];
    const float* hu     = (const float*)d_in[5];
    float* out = (float*)d_out;
    float* ws  = (float*)d_ws;

    fbp_prep<<<B_, 256, 0, stream>>>(angles, dso, ddo, du, hu, ws);

    fbp_toeplitz<<<(DET_ * DET_ + 255) / 256, 256, 0, stream>>>(ws);

    const int total = B_ * A_ * DET_;
    fbp_weight<<<(total + 255) / 256, 256, 0, stream>>>(sino, ws);

    // 23 column-tile pairs covered by 3 blocks x 8 waves; 72 row tiles; B batches
    fbp_filter_wmma<<<dim3(3, A_ / 16, B_), 256, 0, stream>>>(ws);

    fbp_backproject<<<dim3(N_ / 16, N_ / 16, B_), dim3(16, 16), 0, stream>>>(ws, out);
}